// upconv_layer_batch_average_26388279067298
// MI455X (gfx1250) — compile-verified
//
#include <hip/hip_runtime.h>
#include <hip/hip_bf16.h>

// Problem constants (from reference)
#define RAW_NODES 10242
#define NEW_NODES (4 * RAW_NODES - 6)       // 40962
#define DOWN_NODES (NEW_NODES - RAW_NODES)  // 30720
#define BATCH 8
#define CHAN 128
#define NROWS (BATCH * CHAN)                // 1024 (b,c) rows
#define TPB 256

// One block per (b,c) row:
//  1) async-copy the 10242-float source row into LDS (5121 x b64 transfers)
//  2) gather top part:   out[j]        = row[top_idx[j]/7]
//  3) gather down part:  out[RAW + k]  = 0.5*(row[d[2k]/7] + row[d[2k+1]/7])
__global__ __launch_bounds__(TPB) void upconv_gather_kernel(
    const float* __restrict__ x,
    const int*   __restrict__ top_idx,
    const int*   __restrict__ down_idx,
    float*       __restrict__ out)
{
    __shared__ __align__(16) float row[RAW_NODES];

    const int t  = threadIdx.x;
    const int bc = blockIdx.x;  // 0 .. NROWS-1

    const float* __restrict__ src = x   + (size_t)bc * RAW_NODES;
    float*       __restrict__ dst = out + (size_t)bc * NEW_NODES;

    // LDS byte offset of the staging buffer: generic LDS addresses carry the
    // LDS offset in the low 32 bits (aperture tag in the high dword).
    const unsigned lds_base = (unsigned)(size_t)(void*)&row[0];

    // ---- Stage row -> LDS via CDNA5 async global->LDS (ASYNCcnt path) ----
    // RAW_NODES = 10242 floats = 5121 exact B64 transfers; every row base is
    // 8-byte aligned (10242*4 = 40968 is a multiple of 8).
    #pragma unroll 4
    for (int i = t; i < RAW_NODES / 2; i += TPB) {
        const unsigned la = lds_base + (unsigned)i * 8u;
        const float*   g  = src + i * 2;
        asm volatile("global_load_async_to_lds_b64 %0, %1, off"
                     :: "v"(la), "v"(g)
                     : "memory");
    }
    asm volatile("s_wait_asynccnt 0" ::: "memory");
    __syncthreads();

    // ---- Top part: plain gather from LDS, coalesced b32 stores ----
    #pragma unroll 2
    for (int j = t; j < RAW_NODES; j += TPB) {
        const int s = top_idx[j] / 7;   // magic-multiply, cheap VALU
        dst[j] = row[s];
    }

    // ---- Down part: pairwise average, pair indices via one b64 load ----
    const int2* __restrict__ dp = (const int2*)down_idx;
    float* __restrict__ dst2 = dst + RAW_NODES;
    #pragma unroll 2
    for (int k = t; k < DOWN_NODES; k += TPB) {
        const int2 p = dp[k];
        const int s0 = p.x / 7;
        const int s1 = p.y / 7;
        dst2[k] = 0.5f * (row[s0] + row[s1]);
    }
}

extern "C" void kernel_launch(void* const* d_in, const int* in_sizes, int n_in,
                              void* d_out, int out_size, void* d_ws, size_t ws_size,
                              hipStream_t stream) {
    (void)in_sizes; (void)n_in; (void)out_size; (void)d_ws; (void)ws_size;

    const float* x        = (const float*)d_in[0];
    const int*   top_idx  = (const int*)d_in[1];   // harness: integer -> int32
    const int*   down_idx = (const int*)d_in[2];
    float*       out      = (float*)d_out;

    upconv_gather_kernel<<<NROWS, TPB, 0, stream>>>(x, top_idx, down_idx, out);
}